// YoloEval_24026047054152
// MI455X (gfx1250) — compile-verified
//
#include <hip/hip_runtime.h>
#include <stdint.h>

#define NTOT   100800      // 4800 + 19200 + 76800 anchor boxes
#define NCLS   80
#define NEGV   (-1000000000.0f)

// dword boundaries of the concatenated feats (85 dwords per box)
#define DW_L1  408000      // 4800*85
#define DW_L2  2040000     // 24000*85

// ---------------------------------------------------------------------------
// Stage 1: decode boxes + build transposed, thresholded score matrix.
// Each wave async-stages its 32 rows (32*85 dwords) into LDS via
// global_load_async_to_lds_b128, waits on ASYNCcnt, then each lane decodes
// its row from LDS (stride 85 dwords => bank-conflict free on 64 banks).
// ---------------------------------------------------------------------------
__global__ __launch_bounds__(128) void yolo_decode(
    const float* __restrict__ f0, const float* __restrict__ f1,
    const float* __restrict__ f2, const float* __restrict__ ishape,
    float* __restrict__ boxes, float* __restrict__ area,
    float* __restrict__ scoresT)
{
    __shared__ __align__(16) float smem[128 * 85];   // 43,520 bytes

    const int tid  = threadIdx.x;
    const int wave = tid >> 5;
    const int lane = tid & 31;
    const int waveBase = blockIdx.x * 128 + wave * 32;   // N % 32 == 0 -> waves fully valid or fully idle
    if (waveBase >= NTOT) return;

    // ---- async stage: 680 float4 per wave, 16B aligned, never straddles a level
    {
        const unsigned smem_base = (unsigned)(size_t)(&smem[0]);   // low 32 bits of flat ptr == LDS offset
        const unsigned ldsw = smem_base + (unsigned)(wave * 2720 * 4);
        const long long gdw = (long long)waveBase * 85;
        for (int i4 = lane; i4 < 680; i4 += 32) {
            long long g = gdw + (long long)i4 * 4;
            const float* src;
            if (g < DW_L1)      src = f0 + g;
            else if (g < DW_L2) src = f1 + (g - DW_L1);
            else                src = f2 + (g - DW_L2);
            unsigned dst = ldsw + (unsigned)(i4 * 16);
            asm volatile("global_load_async_to_lds_b128 %0, %1, off"
                         :: "v"(dst), "v"(src) : "memory");
        }
        asm volatile("s_wait_asynccnt 0x0" ::: "memory");
    }

    const int n = waveBase + lane;
    const float* row = &smem[(wave * 32 + lane) * 85];

    // ---- level / grid decomposition
    int rel, W, lvl;
    if (n < 4800)       { lvl = 0; rel = n;         W = 40;  }
    else if (n < 24000) { lvl = 1; rel = n - 4800;  W = 80;  }
    else                { lvl = 2; rel = n - 24000; W = 160; }
    const int a    = rel % 3;
    const int cell = rel / 3;
    const int gx   = cell % W;
    const int gy   = cell / W;

    // anchors per level (ANCHOR_MASK applied): [w, h]
    const float AW[3][3] = { {116.f,156.f,373.f}, {30.f,62.f,59.f}, {10.f,16.f,33.f} };
    const float AH[3][3] = { { 90.f,198.f,326.f}, {61.f,45.f,119.f}, {13.f,30.f,23.f} };
    const float aw = AW[lvl][a], ah = AH[lvl][a];

    const float imgH = ishape[0], imgW = ishape[1];
    const float inS  = 1280.0f;                     // 40*32 (square input)
    const float fW   = (float)W;

    const float t0 = row[0], t1 = row[1], t2 = row[2], t3 = row[3], t4 = row[4];
    const float sx = 1.0f / (1.0f + expf(-t0));
    const float sy = 1.0f / (1.0f + expf(-t1));
    const float bx = (sx + (float)gx) / fW;
    const float by = (sy + (float)gy) / fW;         // H == W per level
    const float bw = expf(t2) * aw / inS;
    const float bh = expf(t3) * ah / inS;
    const float conf = 1.0f / (1.0f + expf(-t4));

    // _correct_boxes
    const float mx     = fmaxf(imgH, imgW);
    const float boxedH = inS * (imgH / mx);
    const float boxedW = inS * (imgW / mx);
    const float offH   = (inS - boxedH) * 0.5f;
    const float offW   = (inS - boxedW) * 0.5f;
    const float scH    = imgH / boxedH;
    const float scW    = imgW / boxedW;

    const float yy = (by * inS - offH) * scH;
    const float xx = (bx * inS - offW) * scW;
    const float hh = bh * inS * scH;
    const float ww = bw * inS * scW;

    const float y1 = fminf(fmaxf(yy - hh * 0.5f, 0.0f), imgH);
    const float x1 = fminf(fmaxf(xx - ww * 0.5f, 0.0f), imgW);
    const float y2 = fminf(fmaxf(yy + hh * 0.5f, 0.0f), imgH);
    const float x2 = fminf(fmaxf(xx + ww * 0.5f, 0.0f), imgW);

    float4 bv; bv.x = y1; bv.y = x1; bv.z = y2; bv.w = x2;
    ((float4*)boxes)[n] = bv;
    area[n] = fmaxf(y2 - y1, 0.0f) * fmaxf(x2 - x1, 0.0f);

    #pragma unroll 4
    for (int c = 0; c < NCLS; ++c) {
        const float p = 1.0f / (1.0f + expf(-row[5 + c]));
        const float s = conf * p;
        scoresT[(size_t)c * NTOT + n] = (s >= 0.6f) ? s : NEGV;
    }
}

// ---------------------------------------------------------------------------
// Stage 2: per-class greedy NMS. One block per class; 20 fused
// suppress+argmax passes over the L2-resident score row.
// ---------------------------------------------------------------------------
__global__ __launch_bounds__(256) void yolo_nms(
    const float* __restrict__ boxes, const float* __restrict__ area,
    float* __restrict__ scoresT, float* __restrict__ out)
{
    const int c   = blockIdx.x;
    const int tid = threadIdx.x;
    float* S = scoresT + (size_t)c * NTOT;

    __shared__ float sval[256];
    __shared__ int   sidx[256];
    __shared__ float bc[6];      // y1,x1,y2,x2,area,score of selected box
    __shared__ int   bsel;

    float sy1 = 0.f, sx1 = 0.f, sy2 = 0.f, sx2 = 0.f, sar = 0.f;
    int   sel = -1;
    bool  have = false;

    for (int step = 0; step < 20; ++step) {
        // fused pass: suppress w.r.t. previous selection, track running argmax
        float best = -3.0e38f;
        int   bidx = NTOT;
        for (int n = tid; n < NTOT; n += 256) {
            float s = S[n];
            if (have && s > 0.5f * NEGV) {          // only live entries need IoU work
                if (n == sel) {
                    s = NEGV; S[n] = NEGV;
                } else {
                    float4 b = ((const float4*)boxes)[n];
                    float iy = fminf(sy2, b.z) - fmaxf(sy1, b.x);
                    float ix = fminf(sx2, b.w) - fmaxf(sx1, b.y);
                    float inter = fmaxf(iy, 0.0f) * fmaxf(ix, 0.0f);
                    float iou = inter / (sar + area[n] - inter + 1e-9f);
                    if (iou > 0.5f) { s = NEGV; S[n] = NEGV; }
                }
            }
            if (s > best) { best = s; bidx = n; }   // increasing n + strict '>' => first max
        }
        sval[tid] = best; sidx[tid] = bidx;
        __syncthreads();

        // block argmax, tie-break on lowest index (matches jnp.argmax)
        for (int off = 128; off > 0; off >>= 1) {
            if (tid < off) {
                float ov = sval[tid + off]; int oi = sidx[tid + off];
                float mv = sval[tid];       int mi = sidx[tid];
                if (ov > mv || (ov == mv && oi < mi)) { sval[tid] = ov; sidx[tid] = oi; }
            }
            __syncthreads();
        }

        if (tid == 0) {
            const int   i  = sidx[0];
            const float sc = sval[0];
            float4 b = ((const float4*)boxes)[i];
            bc[0] = b.x; bc[1] = b.y; bc[2] = b.z; bc[3] = b.w;
            bc[4] = area[i]; bc[5] = sc; bsel = i;

            const bool valid = sc > 0.5f * NEGV;
            const int  o = c * 20 + step;
            out[o * 4 + 0] = valid ? (float)(int)b.x : 0.0f;
            out[o * 4 + 1] = valid ? (float)(int)b.y : 0.0f;
            out[o * 4 + 2] = valid ? (float)(int)b.z : 0.0f;
            out[o * 4 + 3] = valid ? (float)(int)b.w : 0.0f;
            out[6400 + o]  = valid ? sc       : 0.0f;
            out[8000 + o]  = valid ? (float)c : -1.0f;
        }
        __syncthreads();
        sy1 = bc[0]; sx1 = bc[1]; sy2 = bc[2]; sx2 = bc[3]; sar = bc[4];
        sel = bsel;
        have = (bc[5] > 0.5f * NEGV);
        __syncthreads();
    }
}

// ---------------------------------------------------------------------------
extern "C" void kernel_launch(void* const* d_in, const int* in_sizes, int n_in,
                              void* d_out, int out_size, void* d_ws, size_t ws_size,
                              hipStream_t stream) {
    const float* f0     = (const float*)d_in[0];   // (1,40,40,3,85)
    const float* f1     = (const float*)d_in[1];   // (1,80,80,3,85)
    const float* f2     = (const float*)d_in[2];   // (1,160,160,3,85)
    const float* ishape = (const float*)d_in[3];   // [768, 1280]

    float* ws      = (float*)d_ws;
    float* boxes   = ws;                       // NTOT*4 floats (16B aligned)
    float* area    = ws + (size_t)NTOT * 4;    // NTOT floats
    float* scoresT = area + NTOT;              // NCLS*NTOT floats (~32 MB)
    float* out     = (float*)d_out;            // 6400 box + 1600 score + 1600 class

    yolo_decode<<<(NTOT + 127) / 128, 128, 0, stream>>>(f0, f1, f2, ishape,
                                                        boxes, area, scoresT);
    yolo_nms<<<NCLS, 256, 0, stream>>>(boxes, area, scoresT, out);
}